// GCNClassifier_58557584114442
// MI455X (gfx1250) — compile-verified
//
#include <hip/hip_runtime.h>
#include <hip/hip_bf16.h>
#include <math.h>

typedef __attribute__((ext_vector_type(2))) float v2f;
typedef __attribute__((ext_vector_type(8))) float v8f;

__device__ __forceinline__ float sigmoidf_(float x) { return 1.0f / (1.0f + expf(-x)); }

// ---------------------------------------------------------------------------
// Generic fp32 WMMA GEMM: C[M x Ntot] = A[M x K] * B^T, where B rows come from
// B0 (rows [0,split)) or B1 (rows [split,Ntot)). A rows optionally gathered
// through an index array (embedding lookup fused into the GEMM).
// Block: 256 threads = 8 waves; block tile 32 rows x 64 cols; each wave owns a
// 16x16 C tile, K consumed in steps of 4 via v_wmma_f32_16x16x4_f32.
// ---------------------------------------------------------------------------
template <bool GATHER, int K>
__global__ __launch_bounds__(256) void wmma_gemm_kernel(
    const float* __restrict__ A, const int* __restrict__ ids,
    const float* __restrict__ table, const float* __restrict__ B0,
    const float* __restrict__ B1, int split, float* __restrict__ C, int ldc,
    int Ntot) {
  __shared__ float a_sh[32][K + 1];
  const int tid = threadIdx.x;
  const int row0 = blockIdx.x * 32;
  const int col0 = blockIdx.y * 64;

  // cooperative (coalesced-ish) stage of 32 A rows into LDS
  for (int idx = tid; idx < 32 * K; idx += 256) {
    int r = idx / K, c = idx - r * K;
    const float* src;
    if (GATHER)
      src = table + (size_t)ids[row0 + r] * K;
    else
      src = A + (size_t)(row0 + r) * K;
    a_sh[r][c] = src[c];
  }
  __syncthreads();

  const int wave = tid >> 5;
  const int lane = tid & 31;
  const int wrow = (wave >> 2) * 16;  // 0 or 16
  const int wcol = col0 + (wave & 3) * 16;
  const int ml = lane & 15;
  const int koff = (lane >> 4) * 2;

  // B fragment source row (clamped for out-of-range columns; store is guarded)
  int bcol = wcol + ml;
  if (bcol >= Ntot) bcol = Ntot - 1;
  const float* Brow = (bcol < split) ? (B0 + (size_t)bcol * K)
                                     : (B1 + (size_t)(bcol - split) * K);

  v8f acc = {0.f, 0.f, 0.f, 0.f, 0.f, 0.f, 0.f, 0.f};
  for (int k = 0; k < K; k += 4) {
    v2f a, b;
    // A 16x4 fp32 layout: lanes 0-15 hold K={0,1}, lanes 16-31 hold K={2,3}
    a.x = a_sh[wrow + ml][k + koff];
    a.y = a_sh[wrow + ml][k + koff + 1];
    // B 4x16 fp32: per-lane column bcol, same K split
    b.x = Brow[k + koff];
    b.y = Brow[k + koff + 1];
    acc = __builtin_amdgcn_wmma_f32_16x16x4_f32(false, a, false, b, (short)0,
                                                acc, false, false);
  }

  // C 16x16 fp32: VGPR v -> M = v + 8*(lane>=16), N = lane&15
  const int mbase = wrow + ((lane >> 4) * 8);
  const int ncol = wcol + ml;
  if (ncol < Ntot) {
    for (int v = 0; v < 8; ++v) {
      int row = row0 + mbase + v;
      C[(size_t)row * ldc + ncol] = acc[v];
    }
  }
}

// ---------------------------------------------------------------------------
// BiLSTM recurrence. grid = (B/16, 2 directions); 16 sentences per block.
// W_hh (200x50) + gate buffer + h/c state resident in LDS (~60 KB < 320 KB).
// xg already holds W_ih @ emb for both directions (cols: dir*200 + gate).
// Gate order (PyTorch): i, f, g, o.
// ---------------------------------------------------------------------------
__global__ __launch_bounds__(256) void bilstm_kernel(
    const float* __restrict__ xg, const float* __restrict__ Whh_f,
    const float* __restrict__ Whh_b, const float* __restrict__ bias_f,
    const float* __restrict__ bias_b, float* __restrict__ x_out, int L) {
  const int dir = blockIdx.y;
  const float* Whh = dir ? Whh_b : Whh_f;
  const float* bias = dir ? bias_b : bias_f;
  const int b0 = blockIdx.x * 16;
  const int tid = threadIdx.x;

  __shared__ float sW[200 * 50];
  __shared__ float sh[16][52];
  __shared__ float sc[16][52];
  __shared__ float sg[16][200];

  for (int i = tid; i < 200 * 50; i += 256) sW[i] = Whh[i];
  for (int i = tid; i < 16 * 50; i += 256) {
    sh[i / 50][i % 50] = 0.f;
    sc[i / 50][i % 50] = 0.f;
  }
  __syncthreads();

  for (int s = 0; s < L; ++s) {
    const int t = dir ? (L - 1 - s) : s;
    for (int idx = tid; idx < 16 * 200; idx += 256) {
      int r = idx / 200, g = idx - r * 200;
      size_t node = (size_t)(b0 + r) * L + t;
      float acc = bias[g] + xg[node * 400 + (size_t)dir * 200 + g];
      const float* w = sW + g * 50;
      const float* h = sh[r];
#pragma unroll 10
      for (int k = 0; k < 50; ++k) acc += h[k] * w[k];
      sg[r][g] = acc;
    }
    __syncthreads();
    for (int idx = tid; idx < 16 * 50; idx += 256) {
      int r = idx / 50, m = idx - r * 50;
      float gi = sg[r][m], gf = sg[r][50 + m], gg = sg[r][100 + m],
            go = sg[r][150 + m];
      float c = sigmoidf_(gf) * sc[r][m] + sigmoidf_(gi) * tanhf(gg);
      float h = sigmoidf_(go) * tanhf(c);
      sc[r][m] = c;
      sh[r][m] = h;
      size_t node = (size_t)(b0 + r) * L + t;
      x_out[node * 100 + (size_t)dir * 50 + m] = h;
    }
    __syncthreads();
  }
}

// ---------------------------------------------------------------------------
// Tree setup: deterministic child lists (<=2 children per node here).
// ---------------------------------------------------------------------------
__global__ void zero_int_kernel(int* p, int n) {
  int i = blockIdx.x * 256 + threadIdx.x;
  if (i < n) p[i] = 0;
}

__global__ void build_children_kernel(const int* __restrict__ parent,
                                      int* __restrict__ ccnt,
                                      int* __restrict__ cidx, int n) {
  int m = blockIdx.x * 256 + threadIdx.x;
  if (m >= n) return;
  int p = parent[m];
  if (p >= 0 && p < n) {
    int s = atomicAdd(&ccnt[p], 1);
    if (s < 2) cidx[p * 2 + s] = m;
  }
}

// ---------------------------------------------------------------------------
// One TreeLSTM level. 4 nodes per block, 64 threads per node.
// Parents GATHER child messages (h_fin doubles as msg_h; msg_fc = f*c).
// IEEE fp add of <=2 children is commutative -> order-independent result.
// ---------------------------------------------------------------------------
__global__ __launch_bounds__(256) void tree_level_kernel(
    const int* __restrict__ depth, const int* __restrict__ nlvl_ptr, int lvl,
    const int* __restrict__ ccnt, const int* __restrict__ cidx,
    const float* __restrict__ iou0h, const float* __restrict__ U_iou_w,
    const float* __restrict__ U_iou_b, const float* __restrict__ b_iou,
    const float* __restrict__ U_f_w, const float* __restrict__ U_f_b,
    const float* __restrict__ H_b, float* __restrict__ h_fin,
    float* __restrict__ msg_fc) {
  if (lvl >= *nlvl_ptr) return;
  const int g = threadIdx.x >> 6;
  const int l = threadIdx.x & 63;
  const int node = blockIdx.x * 4 + g;
  const bool active = (depth[node] == lvl);

  __shared__ int anyact;
  if (threadIdx.x == 0) anyact = 0;
  __syncthreads();
  if (active && l == 0) atomicOr(&anyact, 1);
  __syncthreads();
  if (!anyact) return;

  __shared__ float hsum[4][56];
  __shared__ float iouS[4][152];
  __shared__ float hnew[4][56];

  int nch = active ? ccnt[node] : 0;
  const bool haskid = nch > 0;
  if (nch > 2) nch = 2;

  float fcs = 0.f;
  if (active && l < 50) {
    float hs = 0.f;
    for (int ci = 0; ci < nch; ++ci) {
      int ch = cidx[node * 2 + ci];
      hs += h_fin[(size_t)ch * 50 + l];
      fcs += msg_fc[(size_t)ch * 50 + l];
    }
    hsum[g][l] = hs;
  }
  __syncthreads();

  if (active) {
    for (int j = l; j < 150; j += 64) {
      float v = iou0h[(size_t)node * 200 + j] + b_iou[j];
      if (haskid) {
        float acc = U_iou_b[j];
        const float* w = U_iou_w + j * 50;
        for (int k = 0; k < 50; ++k) acc += hsum[g][k] * w[k];
        v += acc;
      }
      iouS[g][j] = v;
    }
  }
  __syncthreads();

  float cnew = 0.f;
  if (active && l < 50) {
    float gi = iouS[g][l], go = iouS[g][50 + l], gu = iouS[g][100 + l];
    cnew = sigmoidf_(gi) * tanhf(gu) + fcs;
    float hinit = iou0h[(size_t)node * 200 + 150 + l] + H_b[l];
    float h = hinit + sigmoidf_(go) * tanhf(cnew);
    hnew[g][l] = h;
    h_fin[(size_t)node * 50 + l] = h;
  }
  __syncthreads();

  if (active && l < 50) {
    float acc = U_f_b[l];
    const float* w = U_f_w + l * 50;
    for (int k = 0; k < 50; ++k) acc += hnew[g][k] * w[k];
    msg_fc[(size_t)node * 50 + l] = sigmoidf_(acc) * cnew;
  }
}

// ---------------------------------------------------------------------------
// Masked mean-pool over target tokens + linear classifier.
// d_out = logits(B x 3) flat, then pooled(B x 50) flat.
// ---------------------------------------------------------------------------
__global__ __launch_bounds__(64) void pool_classify_kernel(
    const float* __restrict__ hfin, const float* __restrict__ tm,
    const float* __restrict__ Cw, const float* __restrict__ Cb,
    float* __restrict__ out, int Bs, int L) {
  __shared__ float pooled[56];
  const int b = blockIdx.x;
  const int tid = threadIdx.x;
  float tmsum = 0.f;
  for (int l = 0; l < L; ++l) tmsum += tm[(size_t)b * L + l];
  if (tid < 50) {
    float s = 0.f;
    for (int l = 0; l < L; ++l) {
      float w = tm[(size_t)b * L + l];
      s += hfin[((size_t)b * L + l) * 50 + tid] * w;
    }
    float p = s / tmsum;
    pooled[tid] = p;
    out[(size_t)Bs * 3 + (size_t)b * 50 + tid] = p;
  }
  __syncthreads();
  if (tid < 3) {
    float acc = Cb[tid];
    for (int k = 0; k < 50; ++k) acc += pooled[k] * Cw[tid * 50 + k];
    out[(size_t)b * 3 + tid] = acc;
  }
}

// ---------------------------------------------------------------------------
extern "C" void kernel_launch(void* const* d_in, const int* in_sizes, int n_in,
                              void* d_out, int out_size, void* d_ws,
                              size_t ws_size, hipStream_t stream) {
  const int* embed_ids = (const int*)d_in[0];
  const int* parent = (const int*)d_in[1];
  const int* depth = (const int*)d_in[2];
  const float* tmask = (const float*)d_in[3];
  const int* num_levels = (const int*)d_in[4];
  const float* table = (const float*)d_in[5];
  const float* W_ih_f = (const float*)d_in[6];
  const float* W_hh_f = (const float*)d_in[7];
  const float* b_f = (const float*)d_in[8];
  const float* W_ih_b = (const float*)d_in[9];
  const float* W_hh_b = (const float*)d_in[10];
  const float* b_b = (const float*)d_in[11];
  const float* W_iou = (const float*)d_in[12];
  const float* U_iou_w = (const float*)d_in[13];
  const float* U_iou_b = (const float*)d_in[14];
  const float* b_iou = (const float*)d_in[15];
  const float* U_f_w = (const float*)d_in[16];
  const float* U_f_b = (const float*)d_in[17];
  const float* H_w = (const float*)d_in[18];
  const float* H_b = (const float*)d_in[19];
  const float* C_w = (const float*)d_in[20];
  const float* C_b = (const float*)d_in[21];

  const int Nn = in_sizes[1];  // total nodes = B*L
  const int L = 64;
  const int Bs = Nn / L;

  // workspace layout (floats)
  float* ws = (float*)d_ws;
  float* xg = ws;                      // N x 400  (fwd|bwd input gates)
  float* xcat = xg + (size_t)Nn * 400; // N x 100  (hf|hb)
  float* iou0h = xcat + (size_t)Nn * 100;  // N x 200 (iou0 | h_init_pre)
  float* hfin = iou0h + (size_t)Nn * 200;  // N x 50  (also msg_h)
  float* mfc = hfin + (size_t)Nn * 50;     // N x 50  (msg f*c)
  int* ccnt = (int*)(mfc + (size_t)Nn * 50);  // N
  int* cidx = ccnt + Nn;                      // 2N

  // 1) tree child lists (deterministic)
  zero_int_kernel<<<(Nn + 255) / 256, 256, 0, stream>>>(ccnt, Nn);
  build_children_kernel<<<(Nn + 255) / 256, 256, 0, stream>>>(parent, ccnt,
                                                              cidx, Nn);

  // 2) fused embedding-gather + input projection GEMM (K=300, 400 outputs)
  wmma_gemm_kernel<true, 300><<<dim3(Nn / 32, 7), 256, 0, stream>>>(
      nullptr, embed_ids, table, W_ih_f, W_ih_b, 200, xg, 400, 400);

  // 3) BiLSTM recurrence
  bilstm_kernel<<<dim3(Bs / 16, 2), 256, 0, stream>>>(xg, W_hh_f, W_hh_b, b_f,
                                                      b_b, xcat, L);

  // 4) TreeLSTM node pre-projection: [W_iou; H_w] @ x  (K=100, 200 outputs)
  wmma_gemm_kernel<false, 100><<<dim3(Nn / 32, 4), 256, 0, stream>>>(
      xcat, nullptr, nullptr, W_iou, H_w, 150, iou0h, 200, 200);

  // 5) level loop (deepest first); on-device num_levels guard
  for (int lvl = 15; lvl >= 0; --lvl) {
    tree_level_kernel<<<(Nn + 3) / 4, 256, 0, stream>>>(
        depth, num_levels, lvl, ccnt, cidx, iou0h, U_iou_w, U_iou_b, b_iou,
        U_f_w, U_f_b, H_b, hfin, mfc);
  }

  // 6) pool + classify
  pool_classify_kernel<<<Bs, 64, 0, stream>>>(hfin, tmask, C_w, C_b,
                                              (float*)d_out, Bs, L);
}